// Meta_Graph1_40114994545303
// MI455X (gfx1250) — compile-verified
//
#include <hip/hip_runtime.h>
#include <hip/hip_bf16.h>

// Problem constants (match reference)
#define BATCH 256
#define NATTR 32
#define DIM   2048

typedef __attribute__((ext_vector_type(2))) float v2f;
typedef __attribute__((ext_vector_type(8))) float v8f;

// ---------------------------------------------------------------------------
// Kernel 1: masked attribute sum  S[b,d] = sum_{a: label[b,a]>0} feat[b,a,d]
// Streams 64 MiB of attribute_feat once; float4 coalesced loads.
// ---------------------------------------------------------------------------
__global__ __launch_bounds__(256) void masked_sum_kernel(
    const float* __restrict__ feat,     // [B, A, D]
    const int*   __restrict__ label,    // [B, A]
    float*       __restrict__ S)        // [B, D]
{
    const int b  = blockIdx.y;                                // 0..255
    const int d4 = blockIdx.x * blockDim.x + threadIdx.x;     // float4 index, 0..511
    const float4* __restrict__ f4 =
        reinterpret_cast<const float4*>(feat) + (size_t)b * NATTR * (DIM / 4) + d4;

    float4 acc = make_float4(0.f, 0.f, 0.f, 0.f);
    const int* lab = label + b * NATTR;   // uniform per block -> scalar loads
#pragma unroll
    for (int a = 0; a < NATTR; ++a) {
        if (lab[a] > 0) {
            float4 v = f4[a * (DIM / 4)];
            acc.x += v.x; acc.y += v.y; acc.z += v.z; acc.w += v.w;
        }
    }
    reinterpret_cast<float4*>(S)[(size_t)b * (DIM / 4) + d4] = acc;
}

// ---------------------------------------------------------------------------
// Kernel 2: Y = tanh(S @ W + bias) via V_WMMA_F32_16X16X4_F32 (full fp32).
// One wave computes a 16x16 output tile; 8 waves/block share an N-tile of W
// (all 8 waves hit the same W cachelines -> L0/L2 broadcast).
//
// Fragment layouts (wave32, ISA 7.12.2):
//   A (16x4):  lane(l<16): v0=A[l,k], v1=A[l,k+1]; lane(l>=16): K+=2
//   B (4x16):  vgpr i, half h: B[k+2h+i, n0 + (l&15)]
//   C/D:       vgpr i, half h: C[m0 + i + 8h, n0 + (l&15)]
//
// Only 2048 waves exist device-wide (fixed by output tiling), so the loop is
// latency-bound; unroll 8 keeps ~24 loads + 8 WMMAs in flight per wave.
// ---------------------------------------------------------------------------
__global__ __launch_bounds__(256) void gemm_tanh_wmma_kernel(
    const float* __restrict__ S,        // [256, 2048]
    const float* __restrict__ W,        // [2048, 2048]
    const float* __restrict__ bias,     // [2048]
    float*       __restrict__ out)      // [256, 2048]
{
    const int K = DIM, N = DIM;
    const int lane = threadIdx.x & 31;
    const int wave = threadIdx.x >> 5;              // 0..7
    const int half = lane >> 4;                     // 0 or 1
    const int lm   = lane & 15;

    const int n0 = blockIdx.x * 16;                 // N-tile
    const int m0 = (blockIdx.y * 8 + wave) * 16;    // M-tile (one per wave)

    const float* __restrict__ Sa = S + (size_t)(m0 + lm) * K;  // A row for this lane
    const float* __restrict__ Wb = W + n0 + lm;                // B column base

    v8f c = {0.f, 0.f, 0.f, 0.f, 0.f, 0.f, 0.f, 0.f};

#pragma unroll 8
    for (int k = 0; k < K; k += 4) {
        const int ka = k + 2 * half;
        // A fragment: contiguous pair -> single b64 load
        v2f a = *reinterpret_cast<const v2f*>(Sa + ka);
        // B fragment: two rows of W, 16 contiguous floats across lanes
        v2f bfrag;
        bfrag.x = Wb[(size_t)(ka + 0) * N];
        bfrag.y = Wb[(size_t)(ka + 1) * N];
        c = __builtin_amdgcn_wmma_f32_16x16x4_f32(
                /*neg_a=*/false, a, /*neg_b=*/false, bfrag,
                /*c_mod=*/(short)0, c, /*reuse_a=*/false, /*reuse_b=*/false);
    }

    const float bv = bias[n0 + lm];
#pragma unroll
    for (int i = 0; i < 8; ++i) {
        const int row = m0 + i + 8 * half;
        out[(size_t)row * N + n0 + lm] = tanhf(c[i] + bv);
    }
}

// ---------------------------------------------------------------------------
// d_in order (setup_inputs dict): 0=x [B,D] (unused: adj[b,A,A]==0),
// 1=attribute_feat [B,A,D], 2=attribute_label [B,A], 3=W [D,D], 4=b [D]
// ---------------------------------------------------------------------------
extern "C" void kernel_launch(void* const* d_in, const int* in_sizes, int n_in,
                              void* d_out, int out_size, void* d_ws, size_t ws_size,
                              hipStream_t stream) {
    const float* feat  = (const float*)d_in[1];
    const int*   label = (const int*)  d_in[2];
    const float* Wmat  = (const float*)d_in[3];
    const float* bias  = (const float*)d_in[4];
    float*       out   = (float*)d_out;
    float*       Sbuf  = (float*)d_ws;              // [256, 2048] fp32 = 2 MiB

    // Kernel 1: masked reduction over attributes
    dim3 g1((DIM / 4) / 256, BATCH);                // (2, 256)
    masked_sum_kernel<<<g1, 256, 0, stream>>>(feat, label, Sbuf);

    // Kernel 2: WMMA GEMM + tanh epilogue
    dim3 g2(DIM / 16, BATCH / (8 * 16));            // (128, 2)
    gemm_tanh_wmma_kernel<<<g2, 256, 0, stream>>>(Sbuf, Wmat, bias, out);
}